// LSTM_apm_27668179321192
// MI455X (gfx1250) — compile-verified
//
#include <hip/hip_runtime.h>
#include <math.h>

// T=30, I=H=O=1024. 4H = 4096.
#define Tn 30
#define Hn 1024
#define G4 4096   // 4*H

typedef float fvec4 __attribute__((ext_vector_type(4)));
typedef float v2f   __attribute__((ext_vector_type(2)));
typedef float v8f   __attribute__((ext_vector_type(8)));

__device__ __forceinline__ float sigf(float x) { return 1.f / (1.f + expf(-x)); }

// ---------------------------------------------------------------------------
// Init: zero ws floats [30*1024, 34*1024) = hs rows 30,31 + zero-row + c.
// ---------------------------------------------------------------------------
__global__ __launch_bounds__(256) void zero_kernel(float* __restrict__ p) {
    p[blockIdx.x * 256 + threadIdx.x] = 0.f;
}

// ---------------------------------------------------------------------------
// Kernel A: gx[t][r] = W_ih[t][r,:]·x[t,:] + b_ih[t][r] + b_hh[t][r]
// One wave per row (122880 rows). Lane-coalesced float4 loads, NT on weights.
// ---------------------------------------------------------------------------
__global__ __launch_bounds__(256) void gx_kernel(const float* __restrict__ W_ih,
                                                 const float* __restrict__ x,
                                                 const float* __restrict__ b_ih,
                                                 const float* __restrict__ b_hh,
                                                 float* __restrict__ gx) {
    const int wave = (blockIdx.x * blockDim.x + threadIdx.x) >> 5;  // row id
    const int lane = threadIdx.x & 31;
    const float* wrow = W_ih + (size_t)wave * Hn;
    const float* xp   = x + (size_t)(wave >> 12) * Hn;  // wave / 4096 = t
    float acc = 0.f;
#pragma unroll
    for (int k = 0; k < 8; ++k) {
        const int off = k * 128 + lane * 4;
        fvec4 wv = __builtin_nontemporal_load((const fvec4*)(wrow + off));
        fvec4 xv = *(const fvec4*)(xp + off);
        acc += wv.x * xv.x + wv.y * xv.y + wv.z * xv.z + wv.w * xv.w;
    }
#pragma unroll
    for (int m = 16; m >= 1; m >>= 1) acc += __shfl_xor(acc, m, 32);
    if (lane == 0) gx[wave] = acc + b_ih[wave] + b_hh[wave];
}

// ---------------------------------------------------------------------------
// Kernel B (one launch per step t): gates = W_hh[t] @ h_prev + gx[t];
// block handles 2 h-elements: waves 0..3 -> gates i,f,g,o of j0; waves 4..7 -> j1.
// ---------------------------------------------------------------------------
__global__ __launch_bounds__(256) void step_kernel(const float* __restrict__ W_hh_t,
                                                   const float* __restrict__ h_prev,
                                                   const float* __restrict__ gx_t,
                                                   const float* __restrict__ ts,
                                                   float* __restrict__ c,
                                                   float* __restrict__ h_out,
                                                   int t) {
    __shared__ float red[8];
    const int w    = threadIdx.x >> 5;
    const int lane = threadIdx.x & 31;
    const int j    = blockIdx.x * 2 + (w >> 2);
    const int row  = (w & 3) * Hn + j;            // gate*H + j
    const float* wrow = W_hh_t + (size_t)row * Hn;
    float acc = 0.f;
#pragma unroll
    for (int k = 0; k < 8; ++k) {
        const int off = k * 128 + lane * 4;
        fvec4 wv = __builtin_nontemporal_load((const fvec4*)(wrow + off));
        fvec4 hv = *(const fvec4*)(h_prev + off);
        acc += wv.x * hv.x + wv.y * hv.y + wv.z * hv.z + wv.w * hv.w;
    }
#pragma unroll
    for (int m = 16; m >= 1; m >>= 1) acc += __shfl_xor(acc, m, 32);
    if (lane == 0) red[w] = acc + gx_t[row];
    __syncthreads();
    if (threadIdx.x < 2) {
        const int jj = blockIdx.x * 2 + threadIdx.x;
        const int b  = threadIdx.x * 4;
        const float ig = sigf(red[b + 0]);
        const float fg = sigf(red[b + 1]);
        const float gg = tanhf(red[b + 2]);
        const float og = sigf(red[b + 3]);
        const float cn = fg * c[jj] + ig * gg;
        const float hh = og * tanhf(cn);
        float d = 1.f;
        if (t < Tn - 1) {
            const float dt = ts[t + 1] - ts[t];
            d = 2.f / (1.f + expf(dt * (1.f / 60.f)));   // 2*sigmoid(-dt/60)
        }
        c[jj]     = cn * d;   // decayed cell state for next step
        h_out[jj] = hh;       // hs[t][jj]
    }
}

// ---------------------------------------------------------------------------
// Kernel C: out = sigmoid(hs[32x1024] @ W_out.T + b_out), rows 30/31 padded.
// One wave per 16x16 tile; K-loop of V_WMMA_F32_16X16X4_F32.
// A layout (16x4 f32): lane<16 holds {K,K+1}, lane>=16 holds {K+2,K+3}, M=lane&15.
// B[k][n] = W_out[n][k]  -> same per-lane addressing against row-major W_out.
// D: VGPR r -> M = r + (lane>=16 ? 8:0), N = lane&15.
// ---------------------------------------------------------------------------
__global__ __launch_bounds__(256) void out_kernel(const float* __restrict__ hs,
                                                  const float* __restrict__ Wout,
                                                  const float* __restrict__ bout,
                                                  float* __restrict__ out) {
    const int w    = threadIdx.x >> 5;
    const int lane = threadIdx.x & 31;
    const int Nt   = blockIdx.x * 8 + w;   // 64 N-tiles
    const int Mt   = blockIdx.y;           // 2 M-tiles
    const int rw   = lane & 15;
    const int koff = (lane >> 4) * 2;
    const float* Ap = hs   + (size_t)(Mt * 16 + rw) * Hn + koff;
    const float* Bp = Wout + (size_t)(Nt * 16 + rw) * Hn + koff;
    v8f acc = {0.f, 0.f, 0.f, 0.f, 0.f, 0.f, 0.f, 0.f};
#pragma unroll 4
    for (int k = 0; k < 256; ++k) {
        v2f a = *(const v2f*)(Ap + k * 4);
        v2f b = *(const v2f*)(Bp + k * 4);
        acc = __builtin_amdgcn_wmma_f32_16x16x4_f32(false, a, false, b,
                                                    (short)0, acc, false, false);
    }
    const int ng = Nt * 16 + rw;
    const float bb = bout[ng];
    const int mbase = Mt * 16 + ((lane >> 4) * 8);
#pragma unroll
    for (int r = 0; r < 8; ++r) {
        const int mg = mbase + r;
        if (mg < Tn) out[mg * Hn + ng] = sigf(acc[r] + bb);
    }
}

// ---------------------------------------------------------------------------
// Launch
// ---------------------------------------------------------------------------
extern "C" void kernel_launch(void* const* d_in, const int* in_sizes, int n_in,
                              void* d_out, int out_size, void* d_ws, size_t ws_size,
                              hipStream_t stream) {
    const float* x     = (const float*)d_in[0];
    const float* ts    = (const float*)d_in[1];
    const float* W_ih  = (const float*)d_in[2];
    const float* W_hh  = (const float*)d_in[3];
    const float* b_ih  = (const float*)d_in[4];
    const float* b_hh  = (const float*)d_in[5];
    const float* W_out = (const float*)d_in[6];
    const float* b_out = (const float*)d_in[7];
    float* out = (float*)d_out;

    float* ws   = (float*)d_ws;
    float* hs   = ws;                 // 32 rows x 1024 (rows 30,31 stay zero)
    float* zrow = ws + 32 * Hn;       // zero h_{-1}
    float* cbuf = ws + 33 * Hn;       // persistent cell state
    float* gx   = ws + 34 * Hn;       // [30][4096]

    // Zero hs rows 30,31 + zrow + cbuf (contiguous 4096 floats).
    zero_kernel<<<16, 256, 0, stream>>>(ws + 30 * Hn);

    // Precompute input-gate contributions for all steps (bulk of bandwidth).
    gx_kernel<<<(Tn * G4) / 8, 256, 0, stream>>>(W_ih, x, b_ih, b_hh, gx);

    // Sequential recurrence.
    for (int t = 0; t < Tn; ++t) {
        const float* hp = (t == 0) ? zrow : (hs + (size_t)(t - 1) * Hn);
        step_kernel<<<Hn / 2, 256, 0, stream>>>(W_hh + (size_t)t * G4 * Hn, hp,
                                                gx + (size_t)t * G4, ts,
                                                cbuf, hs + (size_t)t * Hn, t);
    }

    // Final projection via f32 WMMA.
    out_kernel<<<dim3(8, 2), 256, 0, stream>>>(hs, W_out, b_out, out);
}